// SimplifiedYOLOv1Loss_6597069767295
// MI455X (gfx1250) — compile-verified
//
#include <hip/hip_runtime.h>
#include <hip/hip_bf16.h>

#define GRID_NUM   7
#define IMG_SIZE   448.0f
#define CELL_SZ    64.0f
#define LCOORD     5.0f
#define LNOOBJ     0.1f
#define EPSV       1e-12f

typedef __attribute__((ext_vector_type(2))) float v2f;
typedef __attribute__((ext_vector_type(8))) float v8f;

// Faithful port of _iou (including the y1_t = cy_t + w_t/2 typo).
__device__ __forceinline__ float iou_cell(float px, float py, float pw, float ph,
                                          float tx, float ty, float tw, float th,
                                          float fi, float fj) {
    float cxp = px * CELL_SZ + fj * CELL_SZ;
    float cyp = py * CELL_SZ + fi * CELL_SZ;
    float wp  = pw * IMG_SIZE, hp = ph * IMG_SIZE;
    float x0p = cxp - 0.5f * wp, x1p = cxp + 0.5f * wp;
    float y0p = cyp - 0.5f * hp, y1p = cyp + 0.5f * hp;

    float cxt = tx * CELL_SZ + fj * CELL_SZ;
    float cyt = ty * CELL_SZ + fi * CELL_SZ;
    float wt  = tw * IMG_SIZE, ht = th * IMG_SIZE;
    float x0t = cxt - 0.5f * wt, x1t = cxt + 0.5f * wt;
    float y0t = cyt - 0.5f * ht;
    float y1t = cyt + 0.5f * wt;   // intentional: reference uses w_true here

    float ux0 = fmaxf(x0p, x0t), ux1 = fminf(x1p, x1t);
    float uy0 = fmaxf(y0p, y0t), uy1 = fminf(y1p, y1t);
    bool  valid = (ux0 < ux1) && (uy0 < uy1);
    float au = (ux1 - ux0) * (uy1 - uy0);
    float ap = (x1p - x0p) * (y1p - y0p);
    float at = (x1t - x0t) * (y1t - y0t);
    float res = au / (ap + at - au + EPSV);
    return valid ? res : 0.0f;
}

// Reduce 256 floats in LDS to one scalar using V_WMMA_F32_16X16X4_F32.
// Must be called by all 32 lanes of wave 0 (EXEC all ones for WMMA).
// A = ones(16x4); each of 4 accumulating WMMAs sums one 64-value B tile into
// column totals; D-layout puts colsum[L%16] in acc[0] of lane L, so a full
// 32-lane xor-reduction of acc[0] yields 2*total.
__device__ __forceinline__ float wmma_reduce_256(const float* __restrict__ s) {
    int lane = (int)(threadIdx.x & 31u);
    v2f a; a.x = 1.0f; a.y = 1.0f;
    v8f acc = {};
#pragma unroll
    for (int i = 0; i < 4; ++i) {
        v2f b;
        b.x = s[i * 64 + lane];
        b.y = s[i * 64 + 32 + lane];
        acc = __builtin_amdgcn_wmma_f32_16x16x4_f32(
            /*neg_a=*/false, a, /*neg_b=*/false, b,
            /*c_mod=*/(short)0, acc, /*reuse_a=*/false, /*reuse_b=*/false);
    }
    float v = acc[0];
#pragma unroll
    for (int off = 16; off > 0; off >>= 1)
        v += __shfl_xor(v, off, 32);
    return 0.5f * v;
}

__global__ void yolo_loss_partial(const float* __restrict__ y_pre,
                                  const float* __restrict__ y_true,
                                  float* __restrict__ partial,
                                  int ncells) {
    __shared__ float smem[256];
    float acc = 0.0f;
    int stride = (int)(gridDim.x * blockDim.x);
    for (int c = (int)(blockIdx.x * blockDim.x + threadIdx.x); c < ncells; c += stride) {
        const v2f* p2 = (const v2f*)(y_pre + (size_t)c * 10);
        v2f p01 = __builtin_nontemporal_load(p2 + 0);   // p0 p1
        v2f p23 = __builtin_nontemporal_load(p2 + 1);   // p2 p3
        v2f p45 = __builtin_nontemporal_load(p2 + 2);   // p4 p5
        v2f p67 = __builtin_nontemporal_load(p2 + 3);   // p6 p7
        v2f p89 = __builtin_nontemporal_load(p2 + 4);   // p8 p9
        const float* t = y_true + (size_t)c * 5;
        float t0 = __builtin_nontemporal_load(t + 0);
        float t1 = __builtin_nontemporal_load(t + 1);
        float t2 = __builtin_nontemporal_load(t + 2);
        float t3 = __builtin_nontemporal_load(t + 3);
        float t4 = __builtin_nontemporal_load(t + 4);

        int cell = c % 49;
        float fi = (float)(cell / 7);
        float fj = (float)(cell % 7);

        float iou0 = iou_cell(p01.x, p01.y, p23.x, p23.y, t0, t1, t2, t3, fi, fj);
        float iou1 = iou_cell(p45.y, p67.x, p67.y, p89.x, t0, t1, t2, t3, fi, fj);

        bool obj = (t4 == 1.0f);
        bool ch0 = iou0 > iou1;

        // obj terms (branchless; selected at the end)
        float cp = ch0 ? p45.x : p89.y;
        float ct = ch0 ? iou0  : iou1;
        float dconf = cp - ct;
        float xp = ch0 ? p01.x : p45.y;
        float yp = ch0 ? p01.y : p67.x;
        float dx = xp - t0, dy = yp - t1;
        float wpre = fmaxf(ch0 ? p23.x : p67.y, EPSV);
        float hpre = fmaxf(ch0 ? p23.y : p89.x, EPSV);
        float wtr  = fmaxf(t2, EPSV), htr = fmaxf(t3, EPSV);
        float dw = __builtin_sqrtf(wpre) - __builtin_sqrtf(wtr);
        float dh = __builtin_sqrtf(hpre) - __builtin_sqrtf(htr);
        float obj_loss = dconf * dconf
                       + LCOORD * (dx * dx + dy * dy)
                       + LCOORD * (dw * dw + dh * dh);
        float noobj_loss = LNOOBJ * (p45.x * p45.x + p89.y * p89.y);

        acc += obj ? obj_loss : noobj_loss;
    }
    smem[threadIdx.x] = acc;
    __syncthreads();
    if (threadIdx.x < 32) {
        float v = wmma_reduce_256(smem);
        if (threadIdx.x == 0) partial[blockIdx.x] = v;
    }
}

__global__ void yolo_loss_final(const float* __restrict__ partial, int n,
                                float* __restrict__ out, float invB) {
    __shared__ float smem[256];
    float acc = 0.0f;
    for (int i = (int)threadIdx.x; i < n; i += 256) acc += partial[i];
    smem[threadIdx.x] = acc;
    __syncthreads();
    if (threadIdx.x < 32) {
        float v = wmma_reduce_256(smem);
        if (threadIdx.x == 0) out[0] = v * invB;
    }
}

extern "C" void kernel_launch(void* const* d_in, const int* in_sizes, int n_in,
                              void* d_out, int out_size, void* d_ws, size_t ws_size,
                              hipStream_t stream) {
    const float* y_pre  = (const float*)d_in[0];
    const float* y_true = (const float*)d_in[1];
    float* out = (float*)d_out;

    int total_pre = in_sizes[0];            // B*7*7*10
    int ncells = total_pre / 10;            // B*49
    int B = total_pre / (GRID_NUM * GRID_NUM * 10);
    float invB = 1.0f / (float)B;

    int nblocks = 2048;
    int max_by_ws = (int)(ws_size / sizeof(float));
    if (nblocks > max_by_ws) nblocks = max_by_ws;
    if (nblocks < 1) nblocks = 1;

    float* partial = (float*)d_ws;
    yolo_loss_partial<<<nblocks, 256, 0, stream>>>(y_pre, y_true, partial, ncells);
    yolo_loss_final<<<1, 256, 0, stream>>>(partial, nblocks, out, invB);
}